// LinearAutoregressiveHMM_34419867910695
// MI455X (gfx1250) — compile-verified
//
#include <hip/hip_runtime.h>
#include <math.h>

typedef float v2f __attribute__((ext_vector_type(2)));
typedef float v8f __attribute__((ext_vector_type(8)));
typedef int vi4 __attribute__((__vector_size__(16)));

#define LOG_2PI 1.8378770664093453f

// ---------------------------------------------------------------------------
// CDNA5 async global->LDS staging (ASYNCcnt-tracked), with fallback.
// ---------------------------------------------------------------------------
#if defined(__gfx1250__) && __has_builtin(__builtin_amdgcn_global_load_async_to_lds_b128)
#define ASYNC_LDS 1
__device__ __forceinline__ void stage_async16(const float* g, float* l) {
    __builtin_amdgcn_global_load_async_to_lds_b128((vi4*)g, (vi4*)l, 0, 0);
}
__device__ __forceinline__ void stage_wait() {
#if __has_builtin(__builtin_amdgcn_s_wait_asynccnt)
    __builtin_amdgcn_s_wait_asynccnt(0);
#else
    asm volatile("s_wait_asynccnt 0x0" ::: "memory");
#endif
}
#else
#define ASYNC_LDS 0
#endif

// ---------------------------------------------------------------------------
// P1: per-state (k = blockIdx.x) Cholesky of (tril@trilT + 1e-6 I), Linv,
//     logdet_half, mu' = Linv@mean.  Block 0 also computes log_trans,
//     log_init and the next-state lookup table.  64 threads per block.
// ---------------------------------------------------------------------------
__global__ __launch_bounds__(64) void prep_kernel(
    const float* __restrict__ CC, const float* __restrict__ T,
    const float* __restrict__ I, const float* __restrict__ Mn,
    float* __restrict__ LinvWS, float* __restrict__ muP,
    float* __restrict__ ldWS, float* __restrict__ lt,
    float* __restrict__ li, int* __restrict__ ns)
{
    __shared__ float A[64 * 65];   // covar -> chol (lower)
    __shared__ float Tl[64 * 65];  // tril  -> Linv
    const int k = blockIdx.x;
    const int r = threadIdx.x;

    for (int c = 0; c < 64; ++c) {
        float v = CC[k * 4096 + r * 64 + c];
        if (c > r) v = 0.0f;
        if (c == r) v = expf(v);
        Tl[r * 65 + c] = v;
    }
    __syncthreads();

    for (int c = 0; c < 64; ++c) {
        int lim = (c < r) ? c : r;
        float s = 0.0f;
        for (int p = 0; p <= lim; ++p) s += Tl[r * 65 + p] * Tl[c * 65 + p];
        A[r * 65 + c] = s + ((c == r) ? 1e-6f : 0.0f);
    }
    __syncthreads();

    for (int j = 0; j < 64; ++j) {
        if (r == j) A[j * 65 + j] = sqrtf(A[j * 65 + j]);
        __syncthreads();
        if (r > j) A[r * 65 + j] /= A[j * 65 + j];
        __syncthreads();
        for (int c = j + 1; c <= r; ++c)
            A[r * 65 + c] -= A[r * 65 + j] * A[c * 65 + j];
        __syncthreads();
    }
    if (r == 0) {
        float s = 0.0f;
        for (int j = 0; j < 64; ++j) s += logf(A[j * 65 + j]);
        ldWS[k] = s;
    }

    {   // Linv column c = r of L X = I
        const int c = r;
        for (int i = 0; i < c; ++i) Tl[i * 65 + c] = 0.0f;
        for (int i = c; i < 64; ++i) {
            float s = (i == c) ? 1.0f : 0.0f;
            for (int p = c; p < i; ++p) s -= A[i * 65 + p] * Tl[p * 65 + c];
            Tl[i * 65 + c] = s / A[i * 65 + i];
        }
        for (int i = 0; i < 64; ++i)
            LinvWS[k * 4096 + i * 64 + c] = Tl[i * 65 + c];
    }
    __syncthreads();

    {   // mu' row r
        float s = 0.0f;
        for (int c = 0; c < 64; ++c) s += Tl[r * 65 + c] * Mn[k * 64 + c];
        muP[k * 64 + r] = s;
    }

    if (k == 0 && r < 8) {
        const int i = r;
        float row[8], mx = -3.0e38f;
        for (int j = 0; j < 8; ++j) { row[j] = T[i * 8 + j]; mx = fmaxf(mx, row[j]); }
        float s = 0.0f;
        for (int j = 0; j < 8; ++j) s += expf(row[j] - mx);
        int arg = 0; float bv = row[0];
        for (int j = 0; j < 8; ++j) {
            float p = expf(row[j] - mx) / s;
            lt[i * 8 + j] = logf(p + 1e-8f);
            if (row[j] > bv) { bv = row[j]; arg = j; }
        }
        ns[i] = arg;
        float m2 = -3.0e38f;
        for (int j = 0; j < 8; ++j) m2 = fmaxf(m2, I[j]);
        float s2 = 0.0f;
        for (int j = 0; j < 8; ++j) s2 += expf(I[j] - m2);
        li[i] = I[i] - (m2 + logf(s2));
    }
}

// ---------------------------------------------------------------------------
// P2: Gt[f][k*64+c] = -(Linv_k @ W_k)[c][f]   for f < 4096
//     Gt[4096+c'][k*64+c] =  Linv_k[c][c']
// ---------------------------------------------------------------------------
__global__ __launch_bounds__(256) void make_gt(
    const float* __restrict__ W, const float* __restrict__ LinvWS,
    float* __restrict__ Gt)
{
    __shared__ float L[4096];
    const int k = blockIdx.y;
    const int tile = blockIdx.x;
    const int tid = threadIdx.x;
    for (int i = tid; i < 4096; i += 256) L[i] = LinvWS[k * 4096 + i];
    __syncthreads();

    if (tile < 16) {
        const int f = tile * 256 + tid;
        float w[64];
        #pragma unroll
        for (int c2 = 0; c2 < 64; ++c2)
            w[c2] = W[(size_t)(k * 64 + c2) * 4096 + f];
        for (int c = 0; c < 64; ++c) {
            float a = 0.0f;
            #pragma unroll
            for (int c2 = 0; c2 < 64; ++c2) a += L[c * 64 + c2] * w[c2];
            Gt[(size_t)f * 512 + k * 64 + c] = -a;
        }
    } else {
        for (int i = tid; i < 4096; i += 256) {
            int c = i >> 6, c2 = i & 63;
            Gt[(size_t)(4096 + c2) * 512 + k * 64 + c] = L[i];
        }
    }
}

// ---------------------------------------------------------------------------
// P3: Wt[f][k*64+c] = W[k][c][f]
// ---------------------------------------------------------------------------
__global__ __launch_bounds__(256) void make_wt(
    const float* __restrict__ W, float* __restrict__ Wt)
{
    const int cc = blockIdx.y;
    const int f = blockIdx.x * 256 + threadIdx.x;
    Wt[(size_t)f * 512 + cc] = W[(size_t)cc * 4096 + f];
}

// ---------------------------------------------------------------------------
// Main GEMM: M=128 (2 batches x 64 t), N=128 (2 states), K'=4160.
// z = G@xe, elp = -0.5*(C log2pi + ||z - mu'||^2) - logdet.
// fp32 WMMA 16x16x4; double-buffered async LDS staging of B.
// A-operand out-of-window elements read a dedicated zero word (address
// select -> v_cndmask, no EXEC-masked branches).
// ---------------------------------------------------------------------------
#define EM_ZOFF (2 * 64 * 65)

__global__ __launch_bounds__(256) void elp_gemm(
    const float* __restrict__ em, const float* __restrict__ Gt,
    const float* __restrict__ muP, const float* __restrict__ ldWS,
    float* __restrict__ elp)
{
    __shared__ float EM[2 * 64 * 65 + 16];  // emissions (stride 65) + zero pad
    __shared__ float BL[2][32 * 144];       // double-buffered B chunk
    const int tid  = threadIdx.x;
    const int wave = tid >> 5, lane = tid & 31;
    const int l16  = lane & 15, half = lane >> 4;
    const int wm   = wave & 3, wn = wave >> 2;
    const int bBase = blockIdx.x * 2;
    const int nBase = blockIdx.y * 128;

    // per-thread staging coords: 4 x 16B per chunk
    const int ccS = (tid * 4) & 127;
    const int rrS = tid >> 5;            // rows rrS + 8j

    // emissions -> LDS
    for (int i = tid * 4; i < 2 * 64 * 64; i += 1024) {
        const int bb = i >> 12, lin = i & 4095;
        float4 t = *(const float4*)&em[(size_t)(bBase + bb) * 4096 + lin];
        const int base = bb * 4160 + (lin >> 6) * 65 + (lin & 63);
        EM[base] = t.x; EM[base + 1] = t.y; EM[base + 2] = t.z; EM[base + 3] = t.w;
    }
    if (tid == 0) EM[EM_ZOFF] = 0.0f;

    // stage chunk 0 into BL[0]
#if ASYNC_LDS
    #pragma unroll
    for (int j = 0; j < 4; ++j)
        stage_async16(&Gt[(size_t)(rrS + j * 8) * 512 + nBase + ccS],
                      &BL[0][(rrS + j * 8) * 144 + ccS]);
    stage_wait();
#else
    {
        float4 q0 = *(const float4*)&Gt[(size_t)(rrS)      * 512 + nBase + ccS];
        float4 q1 = *(const float4*)&Gt[(size_t)(rrS + 8)  * 512 + nBase + ccS];
        float4 q2 = *(const float4*)&Gt[(size_t)(rrS + 16) * 512 + nBase + ccS];
        float4 q3 = *(const float4*)&Gt[(size_t)(rrS + 24) * 512 + nBase + ccS];
        *(float4*)&BL[0][(rrS)      * 144 + ccS] = q0;
        *(float4*)&BL[0][(rrS + 8)  * 144 + ccS] = q1;
        *(float4*)&BL[0][(rrS + 16) * 144 + ccS] = q2;
        *(float4*)&BL[0][(rrS + 24) * 144 + ccS] = q3;
    }
#endif
    __syncthreads();

    v8f acc[2][4];
    v8f z8 = {0.f, 0.f, 0.f, 0.f, 0.f, 0.f, 0.f, 0.f};
    #pragma unroll
    for (int mi = 0; mi < 2; ++mi)
        #pragma unroll
        for (int nj = 0; nj < 4; ++nj) acc[mi][nj] = z8;

    int cur = 0;
    for (int f0 = 0; f0 < 4160; f0 += 32) {
        const int nxt = cur ^ 1;
        const bool more = (f0 + 32) < 4160;

#if ASYNC_LDS
        if (more) {
            #pragma unroll
            for (int j = 0; j < 4; ++j)
                stage_async16(&Gt[(size_t)(f0 + 32 + rrS + j * 8) * 512 + nBase + ccS],
                              &BL[nxt][(rrS + j * 8) * 144 + ccS]);
        }
#else
        float4 q0, q1, q2, q3;
        if (more) {
            q0 = *(const float4*)&Gt[(size_t)(f0 + 32 + rrS)      * 512 + nBase + ccS];
            q1 = *(const float4*)&Gt[(size_t)(f0 + 32 + rrS + 8)  * 512 + nBase + ccS];
            q2 = *(const float4*)&Gt[(size_t)(f0 + 32 + rrS + 16) * 512 + nBase + ccS];
            q3 = *(const float4*)&Gt[(size_t)(f0 + 32 + rrS + 24) * 512 + nBase + ccS];
        }
#endif
        const float* Bc = &BL[cur][0];
        #pragma unroll
        for (int kk = 0; kk < 32; kk += 4) {
            const int fe0 = f0 + kk + 2 * half;
            v2f a[2];
            #pragma unroll
            for (int mi = 0; mi < 2; ++mi) {
                int m = wm * 32 + mi * 16 + l16;
                int bb = m >> 6, t = m & 63;
                int i0 = t * 64 + fe0 - 4096;
                int a0 = bb * 4160 + (i0 >> 6) * 65 + (i0 & 63);
                a0 = (i0 >= 0) ? a0 : EM_ZOFF;
                int i1 = i0 + 1;
                int a1 = bb * 4160 + (i1 >> 6) * 65 + (i1 & 63);
                a1 = (i1 >= 0) ? a1 : EM_ZOFF;
                a[mi].x = EM[a0];
                a[mi].y = EM[a1];
            }
            #pragma unroll
            for (int nj = 0; nj < 4; ++nj) {
                const int col = wn * 64 + nj * 16 + l16;
                const int kb  = kk + 2 * half;
                v2f b;
                b.x = Bc[kb * 144 + col];
                b.y = Bc[(kb + 1) * 144 + col];
                #pragma unroll
                for (int mi = 0; mi < 2; ++mi)
                    acc[mi][nj] = __builtin_amdgcn_wmma_f32_16x16x4_f32(
                        false, a[mi], false, b, (short)0, acc[mi][nj], false, false);
            }
        }
#if ASYNC_LDS
        if (more) stage_wait();
#else
        if (more) {
            *(float4*)&BL[nxt][(rrS)      * 144 + ccS] = q0;
            *(float4*)&BL[nxt][(rrS + 8)  * 144 + ccS] = q1;
            *(float4*)&BL[nxt][(rrS + 16) * 144 + ccS] = q2;
            *(float4*)&BL[nxt][(rrS + 24) * 144 + ccS] = q3;
        }
#endif
        __syncthreads();
        cur = nxt;
    }

    // epilogue: subtract mu', square, reduce over 64 columns per state
    const int k = blockIdx.y * 2 + wn;
    const float ld = ldWS[k];
    float part[2][8];
    #pragma unroll
    for (int mi = 0; mi < 2; ++mi)
        #pragma unroll
        for (int v = 0; v < 8; ++v) part[mi][v] = 0.0f;

    #pragma unroll
    for (int nj = 0; nj < 4; ++nj) {
        float mu = muP[k * 64 + nj * 16 + l16];
        #pragma unroll
        for (int mi = 0; mi < 2; ++mi)
            #pragma unroll
            for (int v = 0; v < 8; ++v) {
                float x = acc[mi][nj][v] - mu;
                part[mi][v] += x * x;
            }
    }
    #pragma unroll
    for (int mi = 0; mi < 2; ++mi)
        #pragma unroll
        for (int v = 0; v < 8; ++v) {
            float s = part[mi][v];
            s += __shfl_xor(s, 1, 32);
            s += __shfl_xor(s, 2, 32);
            s += __shfl_xor(s, 4, 32);
            s += __shfl_xor(s, 8, 32);
            part[mi][v] = s;
        }
    if (l16 == 0) {
        #pragma unroll
        for (int mi = 0; mi < 2; ++mi)
            #pragma unroll
            for (int v = 0; v < 8; ++v) {
                int m = wm * 32 + mi * 16 + v + 8 * half;
                int b = bBase + (m >> 6), t = m & 63;
                elp[(size_t)(b * 64 + t) * 8 + k] =
                    -0.5f * (64.0f * LOG_2PI + part[mi][v]) - ld;
            }
    }
}

// ---------------------------------------------------------------------------
// Forward algorithm: one thread per batch, K=8, L=64.
// ---------------------------------------------------------------------------
__global__ __launch_bounds__(256) void forward_kernel(
    const float* __restrict__ elp, const float* __restrict__ ltWS,
    const float* __restrict__ liWS, int* __restrict__ st)
{
    __shared__ float LT[64];
    __shared__ float LI[8];
    const int tid = threadIdx.x;
    if (tid < 64) LT[tid] = ltWS[tid];
    if (tid < 8) LI[tid] = liWS[tid];
    __syncthreads();
    const int b = blockIdx.x * 256 + tid;
    if (b >= 512) return;

    float la[8];
    #pragma unroll
    for (int j = 0; j < 8; ++j) la[j] = LI[j] + elp[(size_t)b * 512 + j];
    for (int t = 1; t < 64; ++t) {
        const float* e = &elp[(size_t)b * 512 + t * 8];
        float nw[8];
        #pragma unroll
        for (int j = 0; j < 8; ++j) {
            float mx = -3.0e38f;
            #pragma unroll
            for (int i = 0; i < 8; ++i) mx = fmaxf(mx, la[i] + LT[i * 8 + j]);
            float s = 0.0f;
            #pragma unroll
            for (int i = 0; i < 8; ++i) s += expf(la[i] + LT[i * 8 + j] - mx);
            nw[j] = mx + logf(s) + e[j];
        }
        #pragma unroll
        for (int j = 0; j < 8; ++j) la[j] = nw[j];
    }
    int bs = 0; float bv = la[0];
    #pragma unroll
    for (int j = 1; j < 8; ++j) if (la[j] > bv) { bv = la[j]; bs = j; }
    st[b] = bs;
}

__global__ __launch_bounds__(256) void advance_state(
    int* __restrict__ st, const int* __restrict__ ns)
{
    int b = blockIdx.x * 256 + threadIdx.x;
    if (b < 512) st[b] = ns[st[b]];
}

// ---------------------------------------------------------------------------
// Prediction GEMM (step p): all_ar = window @ Wt, select state, add means.
// M=128 batches x N=128 (k,c); K'=4096; double-buffered A and B staging.
// ---------------------------------------------------------------------------
__global__ __launch_bounds__(256) void pred_gemm(
    const float* __restrict__ Wt, const float* __restrict__ Mn,
    const int* __restrict__ st, const float* __restrict__ ring,
    float* __restrict__ out, int p)
{
    __shared__ float AL[2][128 * 36];
    __shared__ float BL[2][32 * 144];
    const int tid  = threadIdx.x;
    const int wave = tid >> 5, lane = tid & 31;
    const int l16  = lane & 15, half = lane >> 4;
    const int wm   = wave & 3, wn = wave >> 2;
    const int bBase = blockIdx.x * 128;
    const int nBase = blockIdx.y * 128;

    // staging coords
    const int ccS = (tid * 4) & 127;     // B: cols
    const int rrS = tid >> 5;            // B: rows rrS + 8j
    const int ccA = (tid * 4) & 31;      // A: cols
    const int rrA = tid >> 3;            // A: rows rrA + 32j

    // stage chunk 0
    {
        const int slot = p & 63;         // frame j=0 of window
#if ASYNC_LDS
        #pragma unroll
        for (int j = 0; j < 4; ++j) {
            stage_async16(&ring[(size_t)(bBase + rrA + j * 32) * 4096 + slot * 64 + ccA],
                          &AL[0][(rrA + j * 32) * 36 + ccA]);
            stage_async16(&Wt[(size_t)(rrS + j * 8) * 512 + nBase + ccS],
                          &BL[0][(rrS + j * 8) * 144 + ccS]);
        }
        stage_wait();
#else
        float4 a0 = *(const float4*)&ring[(size_t)(bBase + rrA)      * 4096 + slot * 64 + ccA];
        float4 a1 = *(const float4*)&ring[(size_t)(bBase + rrA + 32) * 4096 + slot * 64 + ccA];
        float4 a2 = *(const float4*)&ring[(size_t)(bBase + rrA + 64) * 4096 + slot * 64 + ccA];
        float4 a3 = *(const float4*)&ring[(size_t)(bBase + rrA + 96) * 4096 + slot * 64 + ccA];
        float4 q0 = *(const float4*)&Wt[(size_t)(rrS)      * 512 + nBase + ccS];
        float4 q1 = *(const float4*)&Wt[(size_t)(rrS + 8)  * 512 + nBase + ccS];
        float4 q2 = *(const float4*)&Wt[(size_t)(rrS + 16) * 512 + nBase + ccS];
        float4 q3 = *(const float4*)&Wt[(size_t)(rrS + 24) * 512 + nBase + ccS];
        *(float4*)&AL[0][(rrA)      * 36 + ccA] = a0;
        *(float4*)&AL[0][(rrA + 32) * 36 + ccA] = a1;
        *(float4*)&AL[0][(rrA + 64) * 36 + ccA] = a2;
        *(float4*)&AL[0][(rrA + 96) * 36 + ccA] = a3;
        *(float4*)&BL[0][(rrS)      * 144 + ccS] = q0;
        *(float4*)&BL[0][(rrS + 8)  * 144 + ccS] = q1;
        *(float4*)&BL[0][(rrS + 16) * 144 + ccS] = q2;
        *(float4*)&BL[0][(rrS + 24) * 144 + ccS] = q3;
#endif
    }
    __syncthreads();

    v8f acc[2][4];
    v8f z8 = {0.f, 0.f, 0.f, 0.f, 0.f, 0.f, 0.f, 0.f};
    #pragma unroll
    for (int mi = 0; mi < 2; ++mi)
        #pragma unroll
        for (int nj = 0; nj < 4; ++nj) acc[mi][nj] = z8;

    int cur = 0;
    for (int f0 = 0; f0 < 4096; f0 += 32) {
        const int nxt = cur ^ 1;
        const bool more = (f0 + 32) < 4096;
        const int fn = f0 + 32;
        const int slotn = (p + (fn >> 6)) & 63;
        const int cbn = fn & 63;

#if ASYNC_LDS
        if (more) {
            #pragma unroll
            for (int j = 0; j < 4; ++j) {
                stage_async16(&ring[(size_t)(bBase + rrA + j * 32) * 4096 + slotn * 64 + cbn + ccA],
                              &AL[nxt][(rrA + j * 32) * 36 + ccA]);
                stage_async16(&Wt[(size_t)(fn + rrS + j * 8) * 512 + nBase + ccS],
                              &BL[nxt][(rrS + j * 8) * 144 + ccS]);
            }
        }
#else
        float4 a0, a1, a2, a3, q0, q1, q2, q3;
        if (more) {
            a0 = *(const float4*)&ring[(size_t)(bBase + rrA)      * 4096 + slotn * 64 + cbn + ccA];
            a1 = *(const float4*)&ring[(size_t)(bBase + rrA + 32) * 4096 + slotn * 64 + cbn + ccA];
            a2 = *(const float4*)&ring[(size_t)(bBase + rrA + 64) * 4096 + slotn * 64 + cbn + ccA];
            a3 = *(const float4*)&ring[(size_t)(bBase + rrA + 96) * 4096 + slotn * 64 + cbn + ccA];
            q0 = *(const float4*)&Wt[(size_t)(fn + rrS)      * 512 + nBase + ccS];
            q1 = *(const float4*)&Wt[(size_t)(fn + rrS + 8)  * 512 + nBase + ccS];
            q2 = *(const float4*)&Wt[(size_t)(fn + rrS + 16) * 512 + nBase + ccS];
            q3 = *(const float4*)&Wt[(size_t)(fn + rrS + 24) * 512 + nBase + ccS];
        }
#endif
        const float* Ac = &AL[cur][0];
        const float* Bc = &BL[cur][0];
        #pragma unroll
        for (int kk = 0; kk < 32; kk += 4) {
            const int kb = kk + 2 * half;
            v2f a[2];
            #pragma unroll
            for (int mi = 0; mi < 2; ++mi) {
                int m = wm * 32 + mi * 16 + l16;
                a[mi].x = Ac[m * 36 + kb];
                a[mi].y = Ac[m * 36 + kb + 1];
            }
            #pragma unroll
            for (int nj = 0; nj < 4; ++nj) {
                const int col = wn * 64 + nj * 16 + l16;
                v2f b;
                b.x = Bc[kb * 144 + col];
                b.y = Bc[(kb + 1) * 144 + col];
                #pragma unroll
                for (int mi = 0; mi < 2; ++mi)
                    acc[mi][nj] = __builtin_amdgcn_wmma_f32_16x16x4_f32(
                        false, a[mi], false, b, (short)0, acc[mi][nj], false, false);
            }
        }
#if ASYNC_LDS
        if (more) stage_wait();
#else
        if (more) {
            *(float4*)&AL[nxt][(rrA)      * 36 + ccA] = a0;
            *(float4*)&AL[nxt][(rrA + 32) * 36 + ccA] = a1;
            *(float4*)&AL[nxt][(rrA + 64) * 36 + ccA] = a2;
            *(float4*)&AL[nxt][(rrA + 96) * 36 + ccA] = a3;
            *(float4*)&BL[nxt][(rrS)      * 144 + ccS] = q0;
            *(float4*)&BL[nxt][(rrS + 8)  * 144 + ccS] = q1;
            *(float4*)&BL[nxt][(rrS + 16) * 144 + ccS] = q2;
            *(float4*)&BL[nxt][(rrS + 24) * 144 + ccS] = q3;
        }
#endif
        __syncthreads();
        cur = nxt;
    }

    const int k = blockIdx.y * 2 + wn;
    #pragma unroll
    for (int nj = 0; nj < 4; ++nj) {
        const int c = nj * 16 + l16;
        const float mn = Mn[k * 64 + c];
        #pragma unroll
        for (int mi = 0; mi < 2; ++mi)
            #pragma unroll
            for (int v = 0; v < 8; ++v) {
                int m = wm * 32 + mi * 16 + v + 8 * half;
                int b = bBase + m;
                if (st[b] == k)
                    out[(size_t)(b * 16 + p) * 64 + c] = acc[mi][nj][v] + mn;
            }
    }
}

// commit prediction of step p into the ring buffer (overwrites oldest frame)
__global__ __launch_bounds__(256) void commit_pred(
    const float* __restrict__ out, float* __restrict__ ring, int p)
{
    int i = blockIdx.x * 256 + threadIdx.x;  // 512*64
    int b = i >> 6, c = i & 63;
    ring[(size_t)b * 4096 + (p & 63) * 64 + c] = out[(size_t)(b * 16 + p) * 64 + c];
}

// ---------------------------------------------------------------------------
extern "C" void kernel_launch(void* const* d_in, const int* in_sizes, int n_in,
                              void* d_out, int out_size, void* d_ws, size_t ws_size,
                              hipStream_t stream)
{
    const float* em = (const float*)d_in[0];
    const float* T  = (const float*)d_in[1];
    const float* I  = (const float*)d_in[2];
    const float* Mn = (const float*)d_in[3];
    const float* CC = (const float*)d_in[4];
    const float* W  = (const float*)d_in[5];
    float* out = (float*)d_out;
    float* ws  = (float*)d_ws;

    float* Linv = ws;                       // 32768
    float* muP  = ws + 32768;               // 512
    float* ld   = ws + 33280;               // 8
    float* lt   = ws + 33288;               // 64
    float* li   = ws + 33352;               // 8
    int*   ns   = (int*)(ws + 33360);       // 8
    int*   st   = (int*)(ws + 33368);       // 512
    float* Gt   = ws + 34048;               // 4160*512
    float* Wt   = Gt + 2129920;             // 4096*512
    float* elp  = Wt + 2097152;             // 512*64*8
    float* ring = elp + 262144;             // 512*64*64

    prep_kernel<<<8, 64, 0, stream>>>(CC, T, I, Mn, Linv, muP, ld, lt, li, ns);
    make_gt<<<dim3(17, 8), 256, 0, stream>>>(W, Linv, Gt);
    make_wt<<<dim3(16, 512), 256, 0, stream>>>(W, Wt);
    elp_gemm<<<dim3(256, 4), 256, 0, stream>>>(em, Gt, muP, ld, elp);
    forward_kernel<<<2, 256, 0, stream>>>(elp, lt, li, st);
    (void)hipMemcpyAsync(ring, em, (size_t)512 * 64 * 64 * sizeof(float),
                         hipMemcpyDeviceToDevice, stream);
    for (int p = 0; p < 16; ++p) {
        advance_state<<<2, 256, 0, stream>>>(st, ns);
        pred_gemm<<<dim3(4, 4), 256, 0, stream>>>(Wt, Mn, st, ring, out, p);
        commit_pred<<<128, 256, 0, stream>>>(out, ring, p);
    }
}